// ncm_output_3427383902842
// MI455X (gfx1250) — compile-verified
//
#include <hip/hip_runtime.h>

typedef __attribute__((ext_vector_type(2))) float v2f;
typedef __attribute__((ext_vector_type(8))) float v8f;

static constexpr int   kB   = 4096;
static constexpr int   kIn  = 640;
static constexpr int   kOut = 64;
static constexpr int   kKT  = 128;     // K-chunk of W staged in LDS (64*128*4 = 32 KB)
static constexpr float kEps = 1e-6f;

// Materialize the scrambled W = weight_v.T.reshape(64, 640) row-major into ws.
// w_flat[f] = wv_flat[(f % 64) * 640 + (f / 64)]
__global__ void prep_w_kernel(const float* __restrict__ wv, float* __restrict__ w)
{
    int f = blockIdx.x * blockDim.x + threadIdx.x;
    if (f < kOut * kIn)
        w[f] = wv[(f & (kOut - 1)) * kIn + (f >> 6)];
}

// wn2[j] = ||w_j||^2  (64 rows of 640)
__global__ void prep_wn2_kernel(const float* __restrict__ w, float* __restrict__ wn2)
{
    int j = threadIdx.x;                    // 64 threads
    const float* row = w + j * kIn;
    float s = 0.f;
    for (int k = 0; k < kIn; k += 4) {
        float4 v = *(const float4*)(row + k);
        s = fmaf(v.x, v.x, s); s = fmaf(v.y, v.y, s);
        s = fmaf(v.z, v.z, s); s = fmaf(v.w, v.w, s);
    }
    wn2[j] = s;
}

// Main kernel: each wave computes a 16(M) x 64(N) strip of the distance matrix
// via V_WMMA_F32_16X16X4_F32, fusing row-norm accumulation into the A loads.
__launch_bounds__(256)
__global__ void ncm_wmma_kernel(const float* __restrict__ x,
                                const float* __restrict__ w,
                                const float* __restrict__ wnorm2,
                                const float* __restrict__ temp,
                                float* __restrict__ out)
{
    __shared__ float wlds[kOut * kKT];      // 32 KB K-chunk of W

    const int lane  = threadIdx.x & 31;
    const int wave  = threadIdx.x >> 5;     // 8 waves (wave32)
    const int mrow  = lane & 15;            // M row (A/D) == N col (B/D) within tile
    const int khalf = (lane >> 4) * 2;      // K sub-offset owned by this half-wave
    const int row0  = blockIdx.x * 128 + wave * 16;

    const float* xrow = x + (row0 + mrow) * kIn + khalf;

    v8f acc0 = {}, acc1 = {}, acc2 = {}, acc3 = {};
    float sx2 = 0.f;   // partial  sum x^2        (this half-wave's k's)
    float se2 = 0.f;   // partial  sum (x+eps)^2

    for (int kt = 0; kt < kIn; kt += kKT) {
        __syncthreads();                    // previous chunk fully consumed
        // Stage 64 x kKT chunk of W into LDS, float4 granularity (2048 vec4)
        for (int i = threadIdx.x; i < kOut * kKT / 4; i += 256) {
            int n = i >> 5;                 // kKT/4 = 32 float4 per row
            int c = i & 31;
            *(float4*)(wlds + n * kKT + 4 * c) =
                *(const float4*)(w + n * kIn + kt + 4 * c);
        }
        __syncthreads();

        #pragma unroll 4
        for (int kk = 0; kk < kKT; kk += 4) {
            // A fragment (16x4 fp32): contiguous float2 per lane
            v2f a = *(const v2f*)(xrow + kt + kk);
            float ax = a.x, ay = a.y;
            sx2 = fmaf(ax, ax, fmaf(ay, ay, sx2));
            float ex = ax + kEps, ey = ay + kEps;
            se2 = fmaf(ex, ex, fmaf(ey, ey, se2));

            // B fragments (4x16 fp32) for the four N tiles, from LDS
            v2f b0 = *(const v2f*)(wlds + ( 0 + mrow) * kKT + khalf + kk);
            v2f b1 = *(const v2f*)(wlds + (16 + mrow) * kKT + khalf + kk);
            v2f b2 = *(const v2f*)(wlds + (32 + mrow) * kKT + khalf + kk);
            v2f b3 = *(const v2f*)(wlds + (48 + mrow) * kKT + khalf + kk);

            acc0 = __builtin_amdgcn_wmma_f32_16x16x4_f32(false, a, false, b0,
                                                         (short)0, acc0, false, false);
            acc1 = __builtin_amdgcn_wmma_f32_16x16x4_f32(false, a, false, b1,
                                                         (short)0, acc1, false, false);
            acc2 = __builtin_amdgcn_wmma_f32_16x16x4_f32(false, a, false, b2,
                                                         (short)0, acc2, false, false);
            acc3 = __builtin_amdgcn_wmma_f32_16x16x4_f32(false, a, false, b3,
                                                         (short)0, acc3, false, false);
        }
    }

    // Each half-wave saw k = {0,1} vs {2,3} mod 4 of row (lane&15): combine.
    sx2 += __shfl_xor(sx2, 16, 32);
    se2 += __shfl_xor(se2, 16, 32);
    float s    = sqrtf(se2);                // ||x_row + eps||
    float invs = 1.0f / s;
    float nx2  = sx2 * invs * invs;         // ||x/s||^2

    float wn0  = wnorm2[ 0 + mrow];
    float wn1  = wnorm2[16 + mrow];
    float wn2v = wnorm2[32 + mrow];
    float wn3  = wnorm2[48 + mrow];
    float tscale = temp[0];                 // == -1 in reference
    const int mhi = (lane >> 4) * 8;        // D layout: M = p + 8*(lane>>4)

    #pragma unroll
    for (int p = 0; p < 8; ++p) {
        int m = p + mhi;
        // Per-row norms live in lane == m (duplicated in m+16): bpermute them.
        float nx2m  = __shfl(nx2,  m, 32);
        float invsm = __shfl(invs, m, 32);
        float* orow = out + (row0 + m) * kOut + mrow;

        float d0 = nx2m - 2.0f * acc0[p] * invsm + wn0;
        orow[ 0] = tscale * sqrtf(fmaxf(d0, 0.0f));
        float d1 = nx2m - 2.0f * acc1[p] * invsm + wn1;
        orow[16] = tscale * sqrtf(fmaxf(d1, 0.0f));
        float d2 = nx2m - 2.0f * acc2[p] * invsm + wn2v;
        orow[32] = tscale * sqrtf(fmaxf(d2, 0.0f));
        float d3 = nx2m - 2.0f * acc3[p] * invsm + wn3;
        orow[48] = tscale * sqrtf(fmaxf(d3, 0.0f));
    }
}

extern "C" void kernel_launch(void* const* d_in, const int* in_sizes, int n_in,
                              void* d_out, int out_size, void* d_ws, size_t ws_size,
                              hipStream_t stream)
{
    const float* x    = (const float*)d_in[0];   // [4096, 640]
    const float* wv   = (const float*)d_in[1];   // [64, 640]  (weight_v)
    const float* temp = (const float*)d_in[2];   // [1]
    float* out  = (float*)d_out;                 // [4096, 64]
    float* wbuf = (float*)d_ws;                  // scrambled W: 40960 floats
    float* wn2  = wbuf + kOut * kIn;             // 64 floats

    prep_w_kernel<<<(kOut * kIn + 255) / 256, 256, 0, stream>>>(wv, wbuf);
    prep_wn2_kernel<<<1, 64, 0, stream>>>(wbuf, wn2);
    ncm_wmma_kernel<<<kB / 128, 256, 0, stream>>>(x, wbuf, wn2, temp, out);
}